// TexRender_42872363549156
// MI455X (gfx1250) — compile-verified
//
#include <hip/hip_runtime.h>
#include <math.h>

// Soft texture rasterizer for MI455X (gfx1250, wave32).
//
// Roofline: ~1 GFLOP, ~13MB reads -> far below both 23.3 TB/s HBM and the
// matrix ceilings; latency/VALU bound. Strategy: stage all face data in LDS,
// evaluate the affine (GEMM-shaped) part of rasterization on the WMMA pipe
// (V_WMMA_F32_16X16X4_F32: A = [1,px,py,0] per pixel, B = per-face affine
// coefficients), and keep the exp/log soft-coverage math on VALU/TRANS.
//
// Round-2 refinement: A/B operand construction is branchless (value selects
// instead of exec-mask splits) so the inner face-tile loop has no control-flow
// divergence before the WMMAs, cutting s_*_saveexec/s_cbranch overhead.

#define IMG_H 256
#define IMG_W 256
#define NFACE 256
#define SIGMAINV 7000.0f
#define FEPS 1e-10f

#define FD_WORDS 32   // compact per-face record in global workspace
#define FD_STRIDE 36  // padded LDS stride: 36*n mod 64 distinct for n=0..15
                      // -> conflict-free 16-lane gather (other 16 lanes broadcast)

typedef __attribute__((ext_vector_type(2))) float v2f;
typedef __attribute__((ext_vector_type(8))) float v8f;

#if defined(__gfx1250__) && __has_builtin(__builtin_amdgcn_wmma_f32_16x16x4_f32)
#define USE_WMMA_F32 1
#endif

// -------------------------------------------------------------------------
// Kernel 1: per-face preprocessing (F=256 -> one block).
// Face record layout (FD_WORDS floats):
//  [0..2]  w0 affine coeffs (const, x, y)
//  [3..5]  w1 affine coeffs
//  [6..8]  z  affine coeffs
//  [9]     valid flag (|A|>EPS && normal.z>0)
//  [10..15] screen verts x0,y0,x1,y1,x2,y2
//  [16..24] edges: (ex,ey,1/(|e|^2+EPS)) for edges 0->1, 1->2, 2->0
//  [25..30] u0,v0,u1,v1,u2,v2
//  [31]    pad
// -------------------------------------------------------------------------
__global__ __launch_bounds__(NFACE) void fr_preprocess(
    const float* __restrict__ pts, const int* __restrict__ faces,
    const float* __restrict__ rot, const float* __restrict__ cpos,
    const float* __restrict__ cproj, const float* __restrict__ uv,
    const int* __restrict__ ft, float* __restrict__ fd,
    float* __restrict__ out_normal)
{
    int f = threadIdx.x;
    float R[9];
#pragma unroll
    for (int i = 0; i < 9; ++i) R[i] = rot[i];
    float cx = cpos[0], cy = cpos[1], cz = cpos[2];
    float pj0 = cproj[0], pj1 = cproj[1], pj2 = cproj[2];

    float PX[3], PY[3], PZ[3], SX[3], SY[3];
#pragma unroll
    for (int v = 0; v < 3; ++v) {
        int idx = faces[f * 3 + v];
        float p0 = pts[idx * 3 + 0] - cx;
        float p1 = pts[idx * 3 + 1] - cy;
        float p2 = pts[idx * 3 + 2] - cz;
        float X = R[0] * p0 + R[1] * p1 + R[2] * p2;
        float Y = R[3] * p0 + R[4] * p1 + R[5] * p2;
        float Z = R[6] * p0 + R[7] * p1 + R[8] * p2;
        PX[v] = X; PY[v] = Y; PZ[v] = Z;
        float d = Z * pj2;               // xy3.z
        SX[v] = X * pj0 / d;
        SY[v] = Y * pj1 / d;
    }

    // face normal (camera space) + normalized output
    float e1x = PX[1] - PX[0], e1y = PY[1] - PY[0], e1z = PZ[1] - PZ[0];
    float e2x = PX[2] - PX[0], e2y = PY[2] - PY[0], e2z = PZ[2] - PZ[0];
    float nx = e1y * e2z - e1z * e2y;
    float ny = e1z * e2x - e1x * e2z;
    float nz = e1x * e2y - e1y * e2x;
    float nl = 1.0f / (sqrtf(nx * nx + ny * ny + nz * nz) + FEPS);
    out_normal[f * 3 + 0] = nx * nl;
    out_normal[f * 3 + 1] = ny * nl;
    out_normal[f * 3 + 2] = nz * nl;

    float x0 = SX[0], y0 = SY[0], x1 = SX[1], y1 = SY[1], x2 = SX[2], y2 = SY[2];
    float A = (x1 - x0) * (y2 - y0) - (x2 - x0) * (y1 - y0);
    bool okA = fabsf(A) > FEPS;
    float invA = okA ? 1.0f / A : 0.0f;
    float valid = (okA && (nz > 0.0f)) ? 1.0f : 0.0f;

    // w0(px,py) = invA*((x1-px)(y2-py)-(x2-px)(y1-py)) expanded to affine form
    float w0c = invA * (x1 * y2 - x2 * y1);
    float w0x = invA * (y1 - y2);
    float w0y = invA * (x2 - x1);
    float w1c = invA * (x2 * y0 - x0 * y2);
    float w1x = invA * (y2 - y0);
    float w1y = invA * (x0 - x2);
    float z0 = PZ[0], z1 = PZ[1], z2 = PZ[2];
    float dz0 = z0 - z2, dz1 = z1 - z2;
    float zc = z2 + w0c * dz0 + w1c * dz1;
    float zx = w0x * dz0 + w1x * dz1;
    float zy = w0y * dz0 + w1y * dz1;

    int i0 = ft[f * 3 + 0], i1 = ft[f * 3 + 1], i2 = ft[f * 3 + 2];

    float* o = fd + f * FD_WORDS;
    o[0] = w0c; o[1] = w0x; o[2] = w0y;
    o[3] = w1c; o[4] = w1x; o[5] = w1y;
    o[6] = zc;  o[7] = zx;  o[8] = zy;
    o[9] = valid;
    o[10] = x0; o[11] = y0; o[12] = x1; o[13] = y1; o[14] = x2; o[15] = y2;
    float ex, ey;
    ex = x1 - x0; ey = y1 - y0;
    o[16] = ex; o[17] = ey; o[18] = 1.0f / (ex * ex + ey * ey + FEPS);
    ex = x2 - x1; ey = y2 - y1;
    o[19] = ex; o[20] = ey; o[21] = 1.0f / (ex * ex + ey * ey + FEPS);
    ex = x0 - x2; ey = y0 - y2;
    o[22] = ex; o[23] = ey; o[24] = 1.0f / (ex * ex + ey * ey + FEPS);
    o[25] = uv[i0 * 2 + 0]; o[26] = uv[i0 * 2 + 1];
    o[27] = uv[i1 * 2 + 0]; o[28] = uv[i1 * 2 + 1];
    o[29] = uv[i2 * 2 + 0]; o[30] = uv[i2 * 2 + 1];
    o[31] = 0.0f;
}

__device__ __forceinline__ float edge_d2(float px, float py, float ax, float ay,
                                         float ex, float ey, float il) {
    float t = ((px - ax) * ex + (py - ay) * ey) * il;
    t = fminf(fmaxf(t, 0.0f), 1.0f);
    float dx = px - (ax + t * ex);
    float dy = py - (ay + t * ey);
    return dx * dx + dy * dy;
}

// -------------------------------------------------------------------------
// Kernel 2: rasterize + shade. 256 threads = 8 waves; each wave owns 16
// consecutive pixels of one image row; each wave loops 16 face-tiles of 16.
// Per tile: 3x v_wmma_f32_16x16x4_f32 give w0/w1/z for the 16x16 pixel-face
// block. In the C/D striping, lane l owns face (l&15) for pixel rows
// r+8*(l>>4), r=0..7 -> per-lane LDS gather of that face's edge geometry.
// -------------------------------------------------------------------------
__global__ __launch_bounds__(256) void fr_raster(
    const float* __restrict__ fdg, const float* __restrict__ tex,
    float* __restrict__ out)
{
    __shared__ float fd[NFACE * FD_STRIDE];
    __shared__ float pr[128 * 8];

    int tid = threadIdx.x;
    // Stage all face records (compact global -> padded LDS)
    for (int i = tid; i < NFACE * FD_WORDS; i += 256) {
        int f = i >> 5, j = i & 31;
        fd[f * FD_STRIDE + j] = fdg[i];
    }
    __syncthreads();

    int wave = tid >> 5;
    int lane = tid & 31;
    bool lo  = (lane & 16) == 0;  // lane half: selects K-rows (A/B), pixel rows (C/D)
    int half = lo ? 0 : 1;
    int nsub = lane & 15;         // face column in tile / pixel M for A-matrix

    int pixbase = blockIdx.x * 128 + wave * 16;  // 16 px, same row
    int row  = pixbase >> 8;
    int colb = pixbase & 255;
    const float dx = 2.0f / IMG_W;
    float py  = 1.0f - (row + 0.5f) * (2.0f / IMG_H);
    float pxb = (colb + 0.5f) * (2.0f / IMG_W) - 1.0f;

    // A matrix 16x4: row M = [1, px_M, py, 0]
    // lanes 0-15: (K0,K1) = (1, px); lanes 16-31: (K2,K3) = (py, 0)
    // Built with value selects -> no exec-mask divergence at the WMMA sites.
    v2f a;
    a.x = lo ? 1.0f : py;
    a.y = lo ? (pxb + (float)nsub * dx) : 0.0f;

    float pxr[8];
#pragma unroll
    for (int r = 0; r < 8; ++r) pxr[r] = pxb + (float)(r + 8 * half) * dx;

    float bz[8], bw0[8], bw1[8], bw2[8], ls[8];
    int bf[8];
#pragma unroll
    for (int r = 0; r < 8; ++r) {
        bz[r] = -1e10f; bw0[r] = bw1[r] = bw2[r] = 0.0f; ls[r] = 0.0f; bf[r] = 0;
    }

    for (int t = 0; t < 16; ++t) {          // uniform -> EXEC all-1s at WMMA
        int face = t * 16 + nsub;
        const float* Fp = &fd[face * FD_STRIDE];

        // Unconditional contiguous loads (both halves read the same address ->
        // LDS broadcast), then branchless v_cndmask packing of B operands.
        float f0 = Fp[0], f1 = Fp[1], f2 = Fp[2];
        float f3 = Fp[3], f4 = Fp[4], f5 = Fp[5];
        float f6 = Fp[6], f7 = Fp[7], f8 = Fp[8];
        float w0v[8], w1v[8], zv[8];
#if defined(USE_WMMA_F32)
        // B matrix 4x16, col n = this lane's face coefficients.
        // lanes 0-15 hold K0,K1 = (const, x); lanes 16-31 hold K2,K3 = (y, 0)
        v2f b0, b1, b2;
        b0.x = lo ? f0 : f2;  b0.y = lo ? f1 : 0.0f;
        b1.x = lo ? f3 : f5;  b1.y = lo ? f4 : 0.0f;
        b2.x = lo ? f6 : f8;  b2.y = lo ? f7 : 0.0f;
        v8f cz = {};
        v8f W0 = __builtin_amdgcn_wmma_f32_16x16x4_f32(false, a, false, b0,
                                                       (short)0, cz, false, false);
        v8f W1 = __builtin_amdgcn_wmma_f32_16x16x4_f32(false, a, false, b1,
                                                       (short)0, cz, false, false);
        v8f ZZ = __builtin_amdgcn_wmma_f32_16x16x4_f32(false, a, false, b2,
                                                       (short)0, cz, false, false);
#pragma unroll
        for (int r = 0; r < 8; ++r) { w0v[r] = W0[r]; w1v[r] = W1[r]; zv[r] = ZZ[r]; }
#else
#pragma unroll
        for (int r = 0; r < 8; ++r) {
            w0v[r] = f0 + pxr[r] * f1 + py * f2;
            w1v[r] = f3 + pxr[r] * f4 + py * f5;
            zv[r]  = f6 + pxr[r] * f7 + py * f8;
        }
#endif
        bool vOK = Fp[9] > 0.5f;
        float X0 = Fp[10], Y0 = Fp[11], X1 = Fp[12], Y1 = Fp[13], X2 = Fp[14], Y2 = Fp[15];
        float ex0 = Fp[16], ey0 = Fp[17], il0 = Fp[18];
        float ex1 = Fp[19], ey1 = Fp[20], il1 = Fp[21];
        float ex2 = Fp[22], ey2 = Fp[23], il2 = Fp[24];

#pragma unroll
        for (int r = 0; r < 8; ++r) {
            float w0 = w0v[r], w1 = w1v[r], w2 = 1.0f - w0 - w1, z = zv[r];
            float X = pxr[r], Y = py;
            bool inside = vOK && (w0 >= 0.0f) && (w1 >= 0.0f) && (w2 >= 0.0f);
            bool upd = inside && (z > bz[r]);  // faces increase with t -> first-max wins
            bz[r]  = upd ? z    : bz[r];
            bf[r]  = upd ? face : bf[r];
            bw0[r] = upd ? w0   : bw0[r];
            bw1[r] = upd ? w1   : bw1[r];
            bw2[r] = upd ? w2   : bw2[r];
            float d2 = edge_d2(X, Y, X0, Y0, ex0, ey0, il0);
            d2 = fminf(d2, edge_d2(X, Y, X1, Y1, ex1, ey1, il1));
            d2 = fminf(d2, edge_d2(X, Y, X2, Y2, ex2, ey2, il2));
            d2 = inside ? 0.0f : d2;
            float miss = 1.0f - __expf(-d2 * SIGMAINV);
            miss = fminf(fmaxf(miss, 1e-10f), 1.0f);
            float lm = __logf(miss);
            ls[r] += vOK ? lm : 0.0f;
        }
    }

    // Reduce across the 16 face-lanes (xor tree; argmax keeps smaller fid on tie)
#pragma unroll
    for (int r = 0; r < 8; ++r) {
#pragma unroll
        for (int s = 1; s < 16; s <<= 1) {
            float zo = __shfl_xor(bz[r], s, 32);
            int   fo = __shfl_xor(bf[r], s, 32);
            float a0 = __shfl_xor(bw0[r], s, 32);
            float a1 = __shfl_xor(bw1[r], s, 32);
            float a2 = __shfl_xor(bw2[r], s, 32);
            float lsum = __shfl_xor(ls[r], s, 32);
            bool take = (zo > bz[r]) || ((zo == bz[r]) && (fo < bf[r]));
            bz[r]  = take ? zo : bz[r];
            bf[r]  = take ? fo : bf[r];
            bw0[r] = take ? a0 : bw0[r];
            bw1[r] = take ? a1 : bw1[r];
            bw2[r] = take ? a2 : bw2[r];
            ls[r] += lsum;
        }
    }

    if (nsub == 0) {   // lanes 0 and 16 publish 8 pixels each
#pragma unroll
        for (int r = 0; r < 8; ++r) {
            int pix = wave * 16 + r + 8 * half;
            float hit = (bz[r] > -1e9f) ? 1.0f : 0.0f;
            float* q = &pr[pix * 8];
            q[0] = (float)bf[r];
            q[1] = bw0[r]; q[2] = bw1[r]; q[3] = bw2[r];
            q[4] = hit;
            q[5] = 1.0f - __expf(ls[r]);
        }
    }
    __syncthreads();

    // Fragment shader: one thread per pixel (128 pixels per block)
    if (tid < 128) {
        const float* q = &pr[tid * 8];
        int fidv = (int)q[0];
        float w0 = q[1], w1 = q[2], w2 = q[3], hit = q[4], prob = q[5];
        const float* Fp = &fd[fidv * FD_STRIDE];
        float u = (w0 * Fp[25] + w1 * Fp[27] + w2 * Fp[29]) * hit;
        float v = (w0 * Fp[26] + w1 * Fp[28] + w2 * Fp[30]) * hit;
        // hardmask channel = w0+w1+w2 = 1 exactly when hit, masked otherwise
        float x = u * 1023.0f;
        float y = (1.0f - v) * 1023.0f;
        float x0f = fminf(fmaxf(floorf(x), 0.0f), 1023.0f);
        float y0f = fminf(fmaxf(floorf(y), 0.0f), 1023.0f);
        int X0 = (int)x0f, Y0 = (int)y0f;
        int X1 = min(X0 + 1, 1023), Y1 = min(Y0 + 1, 1023);
        float wx = fminf(fmaxf(x - x0f, 0.0f), 1.0f);
        float wy = fminf(fmaxf(y - y0f, 0.0f), 1.0f);
        int pid = blockIdx.x * 128 + tid;
#pragma unroll
        for (int c = 0; c < 3; ++c) {
            const float* tc = tex + (c << 20);       // 1024*1024 per channel
            float c00 = tc[(Y0 << 10) + X0];
            float c01 = tc[(Y0 << 10) + X1];
            float c10 = tc[(Y1 << 10) + X0];
            float c11 = tc[(Y1 << 10) + X1];
            float col = (c00 * (1.0f - wx) + c01 * wx) * (1.0f - wy) +
                        (c10 * (1.0f - wx) + c11 * wx) * wy;
            col *= hit;
            out[pid * 3 + c] = fminf(fmaxf(col, 0.0f), 1.0f);
        }
        out[IMG_H * IMG_W * 3 + pid] = prob;
    }
}

extern "C" void kernel_launch(void* const* d_in, const int* in_sizes, int n_in,
                              void* d_out, int out_size, void* d_ws, size_t ws_size,
                              hipStream_t stream) {
    const float* pts   = (const float*)d_in[0];  // (1,160,3)
    const int*   faces = (const int*)  d_in[1];  // (256,3)
    const float* rot   = (const float*)d_in[2];  // (1,3,3)
    const float* cpos  = (const float*)d_in[3];  // (1,3)
    const float* cproj = (const float*)d_in[4];  // (3,1)
    const float* uv    = (const float*)d_in[5];  // (1,160,2)
    const int*   ft    = (const int*)  d_in[6];  // (256,3)
    const float* tex   = (const float*)d_in[7];  // (1,3,1024,1024)
    float* out = (float*)d_out;                  // imrender | improb | normal1
    float* fdws = (float*)d_ws;                  // 256*32 floats = 32KB

    float* out_normal = out + IMG_H * IMG_W * 3 + IMG_H * IMG_W; // offset 262144

    fr_preprocess<<<1, NFACE, 0, stream>>>(pts, faces, rot, cpos, cproj, uv, ft,
                                           fdws, out_normal);
    fr_raster<<<(IMG_H * IMG_W) / 128, 256, 0, stream>>>(fdws, tex, out);
}